// GAT_7430293422978
// MI455X (gfx1250) — compile-verified
//
#include <hip/hip_runtime.h>
#include <math.h>

typedef float v2f __attribute__((ext_vector_type(2)));
typedef float v8f __attribute__((ext_vector_type(8)));

#define HIDDEN 64
#define LEAKY 0.01f

// ---------------------------------------------------------------------------
// K1: init per-node scratch (maxkey / degree / fill)
// ---------------------------------------------------------------------------
__global__ void gat_init_kernel(unsigned* __restrict__ maxkey,
                                int* __restrict__ degree,
                                int* __restrict__ fill,
                                int n) {
    int i = blockIdx.x * blockDim.x + threadIdx.x;
    if (i < n) {
        maxkey[i] = 0u;   // below key(-inf) = 0x007FFFFF
        degree[i] = 0;
        fill[i]   = 0;
    }
}

// ---------------------------------------------------------------------------
// K2: s_i = x @ w_i, s_j = x @ w_j via V_WMMA_F32_16X16X4_F32
// One wave handles 16 nodes (M). B-tile columns 0/1 hold w_i/w_j, rest zero.
// B fragments: branch-free selects, hoisted. A fragments: staged into a
// register array first so all 16 b64 loads issue back-to-back, then the
// 16 chained WMMAs run without per-step full loadcnt stalls.
// ---------------------------------------------------------------------------
__global__ void gat_scores_wmma_kernel(const float* __restrict__ x,
                                       const float* __restrict__ w_i,
                                       const float* __restrict__ w_j,
                                       float* __restrict__ s_i,
                                       float* __restrict__ s_j,
                                       int n) {
    int wave = threadIdx.x >> 5;
    int lane = threadIdx.x & 31;
    int node_base = (blockIdx.x * 8 + wave) * 16;
    if (node_base >= n) return;          // wave-uniform exit

    int m  = lane & 15;                  // A-row (M) for lanes 0-15, also B-col (N)
    int hl = lane >> 4;                  // lane-half selects K pair
    int node = node_base + m;
    if (node >= n) node = n - 1;         // clamp: EXEC must stay all-1 for WMMA
    const float* xrow = x + (long)node * HIDDEN;

    bool is0 = (m == 0);
    bool is1 = (m == 1);

    // Pre-build all 16 B fragments, branch-free (cndmask selects, no exec ops)
    v2f bfrag[16];
#pragma unroll
    for (int k = 0; k < 16; ++k) {
        int kk = 4 * k + 2 * hl;
        float wi0 = w_i[kk], wi1 = w_i[kk + 1];
        float wj0 = w_j[kk], wj1 = w_j[kk + 1];
        v2f b;
        b.x = is0 ? wi0 : (is1 ? wj0 : 0.0f);
        b.y = is0 ? wi1 : (is1 ? wj1 : 0.0f);
        bfrag[k] = b;
    }

    // Stage all 16 A fragments (issues 16 independent b64 loads)
    v2f afrag[16];
#pragma unroll
    for (int k = 0; k < 16; ++k) {
        int kk = 4 * k + 2 * hl;
        v2f a;
        a.x = xrow[kk];
        a.y = xrow[kk + 1];
        afrag[k] = a;
    }

    v8f acc = {};
#pragma unroll
    for (int k = 0; k < 16; ++k) {
        acc = __builtin_amdgcn_wmma_f32_16x16x4_f32(
            /*neg_a=*/false, afrag[k], /*neg_b=*/false, bfrag[k],
            /*c_mod=*/(short)0, acc, /*reuse_a=*/false, /*reuse_b=*/false);
    }

    // D layout: VGPR r holds D[M=r+8*hl][N=lane&15]; N==0 -> s_i, N==1 -> s_j
    if (m <= 1) {
        float* dstbuf = is0 ? s_i : s_j;
#pragma unroll
        for (int r = 0; r < 8; ++r) {
            int nd = node_base + r + 8 * hl;
            if (nd < n) dstbuf[nd] = acc[r];
        }
    }
}

// ---------------------------------------------------------------------------
// K3: per-edge score + segment-max (order-preserving uint key) + degree count
// ---------------------------------------------------------------------------
__device__ __forceinline__ unsigned float_order_key(float f) {
    unsigned u = __float_as_uint(f);
    return (u & 0x80000000u) ? ~u : (u | 0x80000000u);
}
__device__ __forceinline__ float float_order_unkey(unsigned k) {
    unsigned u = (k & 0x80000000u) ? (k ^ 0x80000000u) : ~k;
    return __uint_as_float(u);
}

__global__ void gat_edge_score_kernel(const int* __restrict__ edge,  // [2,E]
                                      const float* __restrict__ s_i,
                                      const float* __restrict__ s_j,
                                      float* __restrict__ e_val,
                                      unsigned* __restrict__ maxkey,
                                      int* __restrict__ degree,
                                      int E) {
    int e = blockIdx.x * blockDim.x + threadIdx.x;
    if (e >= E) return;
    int src = edge[e];       // edge_index[0] = j (source / gathered row)
    int dst = edge[E + e];   // edge_index[1] = i (destination / segment id)
    float v = s_i[dst] + s_j[src];
    v = (v >= 0.0f) ? v : LEAKY * v;     // leaky_relu
    e_val[e] = v;
    atomicMax(&maxkey[dst], float_order_key(v));
    atomicAdd(&degree[dst], 1);
}

// ---------------------------------------------------------------------------
// K4: exclusive scan of degree -> rowstart (single 1024-thread workgroup)
// ---------------------------------------------------------------------------
__global__ void gat_scan_kernel(const int* __restrict__ degree,
                                int* __restrict__ rowstart,
                                int n) {
    __shared__ int sh[1024];
    __shared__ int carry_sh;
    int tid = threadIdx.x;
    if (tid == 0) carry_sh = 0;
    __syncthreads();
    for (int base = 0; base < n; base += 1024) {
        int i = base + tid;
        int v = (i < n) ? degree[i] : 0;
        sh[tid] = v;
        __syncthreads();
        for (int off = 1; off < 1024; off <<= 1) {
            int t = (tid >= off) ? sh[tid - off] : 0;
            __syncthreads();
            sh[tid] += t;
            __syncthreads();
        }
        int carry = carry_sh;
        if (i < n) rowstart[i] = carry + sh[tid] - v;   // exclusive
        __syncthreads();
        if (tid == 0) carry_sh = carry + sh[1023];
        __syncthreads();
    }
}

// ---------------------------------------------------------------------------
// K5: ex = exp(e - segmax); CSR fill of packed (ex, src) records
// ---------------------------------------------------------------------------
__global__ void gat_edge_exp_fill_kernel(const int* __restrict__ edge,
                                         const float* __restrict__ e_val,
                                         const unsigned* __restrict__ maxkey,
                                         const int* __restrict__ rowstart,
                                         int* __restrict__ fill,
                                         int2* __restrict__ csr_pack,
                                         int E) {
    int e = blockIdx.x * blockDim.x + threadIdx.x;
    if (e >= E) return;
    int src = edge[e];
    int dst = edge[E + e];
    float mx = float_order_unkey(maxkey[dst]);
    float ex = expf(e_val[e] - mx);
    int pos = rowstart[dst] + atomicAdd(&fill[dst], 1);
    int2 pk;
    pk.x = __float_as_int(ex);
    pk.y = src;
    csr_pack[pos] = pk;                 // single b64 store per edge
}

// ---------------------------------------------------------------------------
// K6: gather-side SpMM. One wave32 per destination node; lane owns channels
// {2*lane, 2*lane+1} -> one float2 (b64) gather per lane covers the full
// 256B row in one wave transaction. One b64 broadcast per CSR record.
// Softmax denominator folded in: out = relu((sum ex*x[src]) / (sum ex)).
// Unroll-by-4 exposes independent record->gather chains for latency hiding.
// ---------------------------------------------------------------------------
__global__ void gat_spmm_kernel(const float* __restrict__ x,
                                const int2* __restrict__ csr_pack,
                                const int* __restrict__ rowstart,
                                const int* __restrict__ degree,
                                float* __restrict__ out,
                                int n) {
    int node = blockIdx.x * 8 + (threadIdx.x >> 5);
    int lane = threadIdx.x & 31;
    if (node >= n) return;          // wave-uniform
    int start = rowstart[node];
    int deg   = degree[node];
    float acc0 = 0.0f, acc1 = 0.0f, den = 0.0f;
#pragma unroll 4
    for (int p = 0; p < deg; ++p) {
        int2 pk = csr_pack[start + p];           // wave-broadcast b64 load
        float a = __int_as_float(pk.x);
        int   s = pk.y;
        const float2 xv = *(const float2*)(x + (long)s * HIDDEN + 2 * lane);
        den  += a;
        acc0 = fmaf(a, xv.x, acc0);
        acc1 = fmaf(a, xv.y, acc1);
    }
    float2 o;
    o.x = 0.0f; o.y = 0.0f;
    if (deg > 0) {
        o.x = fmaxf(acc0 / den, 0.0f);
        o.y = fmaxf(acc1 / den, 0.0f);
    }
    *(float2*)(out + (long)node * HIDDEN + 2 * lane) = o;  // single b64 store
}

// ---------------------------------------------------------------------------
// launch
// ---------------------------------------------------------------------------
extern "C" void kernel_launch(void* const* d_in, const int* in_sizes, int n_in,
                              void* d_out, int out_size, void* d_ws, size_t ws_size,
                              hipStream_t stream) {
    const float* x    = (const float*)d_in[0];
    const int*   edge = (const int*)d_in[1];
    const float* w_i  = (const float*)d_in[2];
    const float* w_j  = (const float*)d_in[3];
    float*       out  = (float*)d_out;

    const int N = in_sizes[0] / HIDDEN;   // 100000
    const int E = in_sizes[1] / 2;        // 1600000

    // workspace layout (4-byte elements): 6N node arrays, E e_val, 2E csr_pack
    float*    ws       = (float*)d_ws;
    float*    s_i      = ws;
    float*    s_j      = ws + (size_t)N;
    unsigned* maxkey   = (unsigned*)(ws + 2 * (size_t)N);
    int*      degree   = (int*)(ws + 3 * (size_t)N);
    int*      rowstart = (int*)(ws + 4 * (size_t)N);
    int*      fillc    = (int*)(ws + 5 * (size_t)N);
    float*    e_val    = ws + 6 * (size_t)N;
    size_t    pk_off   = (6 * (size_t)N + (size_t)E + 1) & ~(size_t)1;  // 8B align
    int2*     csr_pack = (int2*)(ws + pk_off);

    // K1: init
    gat_init_kernel<<<(N + 255) / 256, 256, 0, stream>>>(maxkey, degree, fillc, N);

    // K2: scores via f32 WMMA (128 nodes / block = 8 waves * 16 nodes)
    gat_scores_wmma_kernel<<<(N + 127) / 128, 256, 0, stream>>>(x, w_i, w_j, s_i, s_j, N);

    // K3: per-edge leaky score + segment max + degree
    gat_edge_score_kernel<<<(E + 255) / 256, 256, 0, stream>>>(edge, s_i, s_j,
                                                               e_val, maxkey, degree, E);

    // K4: exclusive scan degree -> rowstart
    gat_scan_kernel<<<1, 1024, 0, stream>>>(degree, rowstart, N);

    // K5: exp + CSR fill (packed records)
    gat_edge_exp_fill_kernel<<<(E + 255) / 256, 256, 0, stream>>>(edge, e_val, maxkey,
                                                                  rowstart, fillc,
                                                                  csr_pack, E);

    // K6: per-node gather SpMM + softmax normalize + relu
    gat_spmm_kernel<<<(N + 7) / 8, 256, 0, stream>>>(x, csr_pack,
                                                     rowstart, degree, out, N);
}